// SelfOccluMask_21431886807426
// MI455X (gfx1250) — compile-verified
//
#include <hip/hip_runtime.h>
#include <stdint.h>

// Problem constants (reference: dispmap (8,1,384,1280) f32, bsline (8,) f32)
constexpr int BATCH  = 8;
constexpr int H      = 384;
constexpr int W      = 1280;
constexpr int TILE_W = 256;   // one column per thread
constexpr int TILE_H = 8;     // rows per block
constexpr int HALO   = 22;    // MAX_DISP + 1
constexpr int MAXC   = TILE_W + 2 * HALO;  // 300 columns max in LDS tile
constexpr int SST    = 304;                // padded stride for S buffer

typedef unsigned int u32x4 __attribute__((ext_vector_type(4)));
typedef int          i32x8 __attribute__((ext_vector_type(8)));
typedef int          i32x4 __attribute__((ext_vector_type(4)));

__device__ __forceinline__ int imind(int a, int b) { return a < b ? a : b; }
__device__ __forceinline__ int imaxd(int a, int b) { return a > b ? a : b; }

__global__ __launch_bounds__(256)
void SelfOccluMask_kernel(const float* __restrict__ disp,
                          const float* __restrict__ bsl,
                          float* __restrict__ out)
{
    const int tid  = threadIdx.x;
    const int col0 = blockIdx.x * TILE_W;
    const int row0 = blockIdx.y * TILE_H;
    const int bb   = blockIdx.z;

    // In-bounds rectangle covering the halo (edge padding == clamp, handled by index math)
    const int col_start = imaxd(col0 - HALO, 0);
    const int col_end   = imind(col0 + TILE_W - 1 + HALO, W - 1);
    const int tw        = col_end - col_start + 1;      // <= 300
    const int row_start = imaxd(row0 - 1, 0);
    const int row_end   = imind(row0 + TILE_H, H - 1);
    const int th        = row_end - row_start + 1;      // <= 10

    __shared__ float s_tile[(TILE_H + 2) * MAXC];       // raw disparity tile (packed, stride tw)
    __shared__ float s_S[TILE_H * SST];                 // vertical 3-row sums

    const float* gsrc = disp + ((size_t)bb * H + row_start) * (size_t)W + col_start;

#if defined(__gfx1250__) && __has_builtin(__builtin_amdgcn_tensor_load_to_lds)
    // --- Tensor Data Mover: DMA the (th x tw) f32 tile (row stride W) into LDS ---
    if (tid < 32) {   // one wave issues the TDM op
        const uint64_t ga  = (uint64_t)(uintptr_t)gsrc;
        const uint32_t lds = (uint32_t)(uintptr_t)(&s_tile[0]);   // low 32 bits = LDS byte offset

        u32x4 g0;
        g0[0] = 1u;                                   // count=1, user descriptor
        g0[1] = lds;                                  // lds_addr
        g0[2] = (uint32_t)ga;                         // global_addr[31:0]
        g0[3] = ((uint32_t)(ga >> 32) & 0x01FFFFFFu)  // global_addr[56:32]
              | (2u << 30);                           // type = 2 ("image")

        const uint32_t td0 = (uint32_t)tw;            // tensor_dim0 (elements)
        const uint32_t td1 = (uint32_t)th;            // tensor_dim1
        i32x8 g1;
        g1[0] = (int)(2u << 16);                      // wg_mask=0, data_size=2 -> 4 bytes
        g1[1] = (int)((td0 & 0xFFFFu) << 16);         // abar_addr=0 | tensor_dim0[15:0]
        g1[2] = (int)(((td0 >> 16) & 0xFFFFu) | ((td1 & 0xFFFFu) << 16));
        g1[3] = (int)(((td1 >> 16) & 0xFFFFu) | ((uint32_t)tw << 16));   // tile_dim0 = tw
        g1[4] = (int)((uint32_t)th & 0xFFFFu);        // tile_dim1 = th, tile_dim2 = 0
        g1[5] = (int)W;                               // tensor_dim0_stride[31:0] = W
        g1[6] = 0;                                    // stride0[47:32]=0, stride1 lo=0
        g1[7] = 0;

        i32x4 g2 = (i32x4){0, 0, 0, 0};               // 2D tile: dims 2..4 unused
        i32x4 g3 = (i32x4){0, 0, 0, 0};
#if __clang_major__ >= 23
        i32x8 g4 = (i32x8){0, 0, 0, 0, 0, 0, 0, 0};
        __builtin_amdgcn_tensor_load_to_lds(g0, g1, g2, g3, g4, 0);
#else
        __builtin_amdgcn_tensor_load_to_lds(g0, g1, g2, g3, 0);
#endif
#if __has_builtin(__builtin_amdgcn_s_wait_tensorcnt)
        __builtin_amdgcn_s_wait_tensorcnt(0);
#endif
    }
    __syncthreads();
#else
    // Fallback (host pass / toolchains without the TDM builtin): cooperative load
    for (int idx = tid; idx < th * tw; idx += 256) {
        const int r = idx / tw;
        const int c = idx - r * tw;
        s_tile[idx] = gsrc[(size_t)r * W + c];
    }
    __syncthreads();
#endif

    // Vertical 3-row sums S(h, x) = d(h-1)+d(h)+d(h+1), rows clamped (== edge pad)
    for (int r = 0; r < TILE_H; ++r) {
        const int h  = row0 + r;
        const int l0 = h - row_start;
        const int lm = imaxd(h - 1, 0) - row_start;
        const int lp = imind(h + 1, H - 1) - row_start;
        for (int c = tid; c < tw; c += 256) {
            s_S[r * SST + c] = s_tile[lm * tw + c] + s_tile[l0 * tw + c] + s_tile[lp * tw + c];
        }
    }
    __syncthreads();

    const float bs = bsl[bb];
    const int   w  = col0 + tid;          // always valid: 1280 % 256 == 0
    const int   cb = w - col_start;       // this thread's column in the S buffer
    const int   cmax = tw - 1;

    float* outp = out + ((size_t)bb * H + row0) * (size_t)W + w;

    if (bs < 0.0f) {
        // maskl: columns w .. w+22, clamped right (edge pad)
        for (int r = 0; r < TILE_H; ++r) {
            const float* Sr = &s_S[r * SST];
            const float baseL = (Sr[cb] + Sr[imind(cb + 1, cmax)]) * (1.0f / 6.0f);
            float m = 1e30f;
#pragma unroll
            for (int i = 0; i < 21; ++i) {
                const float v = fabsf(baseL - Sr[imind(cb + i + 2, cmax)] * (1.0f / 3.0f)
                                      + (float)(i + 1));
                m = fminf(m, v);
            }
            outp[(size_t)r * W] = fminf(fmaxf(m, 0.0f), 1.0f);
        }
    } else if (bs > 0.0f) {
        // maskr == flipped compute: columns w .. w-22, clamped left
        for (int r = 0; r < TILE_H; ++r) {
            const float* Sr = &s_S[r * SST];
            const float baseR = (Sr[cb] + Sr[imaxd(cb - 1, 0)]) * (1.0f / 6.0f);
            float m = 1e30f;
#pragma unroll
            for (int i = 0; i < 21; ++i) {
                const float v = fabsf(baseR - Sr[imaxd(cb - i - 2, 0)] * (1.0f / 3.0f)
                                      + (float)(i + 1));
                m = fminf(m, v);
            }
            outp[(size_t)r * W] = fminf(fmaxf(m, 0.0f), 1.0f);
        }
    } else {
        for (int r = 0; r < TILE_H; ++r) outp[(size_t)r * W] = 0.0f;
    }
}

extern "C" void kernel_launch(void* const* d_in, const int* in_sizes, int n_in,
                              void* d_out, int out_size, void* d_ws, size_t ws_size,
                              hipStream_t stream) {
    const float* disp = (const float*)d_in[0];   // (8,1,384,1280) f32
    const float* bsl  = (const float*)d_in[1];   // (8,) f32
    float* out = (float*)d_out;                  // (8,1,384,1280) f32

    dim3 grid(W / TILE_W, H / TILE_H, BATCH);    // (5, 48, 8)
    dim3 block(256);                             // 8 wave32
    SelfOccluMask_kernel<<<grid, block, 0, stream>>>(disp, bsl, out);
}